// GAT_84507776516243
// MI455X (gfx1250) — compile-verified
//
#include <hip/hip_runtime.h>

// ---------------------------------------------------------------------------
// GATv2 x3 (+ dead 4th layer skipped: reference sets h4 = h3) + pooling + MLP
// MI455X / gfx1250: wave32. Dense transforms on V_WMMA_F32_16X16X4_F32 (exact
// f32 math). Edge softmax/aggregation = L2-resident edge-parallel atomics
// (all node tensors << 192MB L2, so random src/dst access never hits HBM).
// ---------------------------------------------------------------------------

typedef float v2f __attribute__((ext_vector_type(2)));
typedef float v8f __attribute__((ext_vector_type(8)));

#define NEG_SLOPE 0.2f
#define BN_EPS    1e-5f
#define NGRAPH    512
#define POOL_COLS 256

// ------------------------- WMMA f32 GEMM ------------------------------------
// C0 = A @ B0 and (if HAS_B1) C1 = A @ B1, sharing A-tile fetches.
// A:[M,K] row-major, B:[K,N] row-major, C:[M,N]. M%16==0, N%(16*NT)==0, K%4==0.
// One wave per 16 x (16*NT) output strip; NT accumulators amortize A loads.
// VGPR layouts per CDNA5 ISA 7.12.2:
//   A 16x4 f32 : lane&15 = M row; VGPR j holds K = 2*(lane>>4) + j
//   B 4x16 f32 : lane&15 = N col; VGPR j holds K = 2*(lane>>4) + j
//   C 16x16 f32: lane&15 = N col; VGPR r holds M = r + 8*(lane>>4)
template<int NT, bool HAS_B1>
__global__ __launch_bounds__(256)
void wmma_gemm(const float* __restrict__ A,
               const float* __restrict__ B0, const float* __restrict__ B1,
               float* __restrict__ C0, float* __restrict__ C1,
               int M, int K, int N)
{
    const int lane   = threadIdx.x & 31;
    const int gw     = (blockIdx.x * blockDim.x + threadIdx.x) >> 5;
    const int tilesN = N / (16 * NT);
    const int tm     = gw / tilesN;
    const int tnB    = (gw - tm * tilesN) * NT;
    if (tm * 16 >= M) return;            // wave-uniform exit: EXEC stays all-1s

    const int  l15  = lane & 15;
    const int  hi   = lane >> 4;         // 0 or 1
    const int  koff = hi * 2;
    const long arow = (long)(tm * 16 + l15) * K;

    int ncol[NT];
#pragma unroll
    for (int t = 0; t < NT; ++t) ncol[t] = (tnB + t) * 16 + l15;

    v8f acc0[NT];
    v8f acc1[NT];
#pragma unroll
    for (int t = 0; t < NT; ++t) { acc0[t] = {}; if (HAS_B1) acc1[t] = {}; }

    for (int k = 0; k < K; k += 4) {
        v2f a;
        a.x = A[arow + k + koff];
        a.y = A[arow + k + koff + 1];
        const float* r0 = B0 + (long)(k + koff)     * N;
        const float* r1 = B0 + (long)(k + koff + 1) * N;
#pragma unroll
        for (int t = 0; t < NT; ++t) {
            v2f b; b.x = r0[ncol[t]]; b.y = r1[ncol[t]];
            acc0[t] = __builtin_amdgcn_wmma_f32_16x16x4_f32(
                          false, a, false, b, (short)0, acc0[t], false, false);
        }
        if (HAS_B1) {
            const float* q0 = B1 + (long)(k + koff)     * N;
            const float* q1 = B1 + (long)(k + koff + 1) * N;
#pragma unroll
            for (int t = 0; t < NT; ++t) {
                v2f b; b.x = q0[ncol[t]]; b.y = q1[ncol[t]];
                acc1[t] = __builtin_amdgcn_wmma_f32_16x16x4_f32(
                              false, a, false, b, (short)0, acc1[t], false, false);
            }
        }
    }
    const int rbase = tm * 16 + hi * 8;
#pragma unroll
    for (int t = 0; t < NT; ++t)
#pragma unroll
        for (int r = 0; r < 8; ++r) {
            C0[(long)(rbase + r) * N + ncol[t]] = acc0[t][r];
            if (HAS_B1) C1[(long)(rbase + r) * N + ncol[t]] = acc1[t][r];
        }
}

// ------------------------- helpers ------------------------------------------
__device__ __forceinline__ unsigned f2key(float f) {
    unsigned b = __float_as_uint(f);
    return (b & 0x80000000u) ? ~b : (b | 0x80000000u);   // monotone float->uint
}
__device__ __forceinline__ float key2f(unsigned u) {
    return (u & 0x80000000u) ? __uint_as_float(u & 0x7FFFFFFFu)
                             : __uint_as_float(~u);
}

__global__ void fill_u32(unsigned* p, unsigned v, long n) {
    long i = (long)blockIdx.x * blockDim.x + threadIdx.x;
    if (i < n) p[i] = v;
}
__global__ void fill_f32(float* p, float v, long n) {
    long i = (long)blockIdx.x * blockDim.x + threadIdx.x;
    if (i < n) p[i] = v;
}
template<int F>
__global__ void init_bias(float* __restrict__ h, const float* __restrict__ b,
                          long total) {
    long i = (long)blockIdx.x * blockDim.x + threadIdx.x;
    if (i < total) h[i] = b[(int)(i & (F - 1))];
}

// ------------------------- edge phase ---------------------------------------
// pass 1: score[e] = leakyrelu(xl[src]+xr[dst]) . a ; segment max via atomicMax
template<int F>
__global__ void edge_score_max(const int* __restrict__ src, const int* __restrict__ dst,
                               int E, int N,
                               const float* __restrict__ xl, const float* __restrict__ xr,
                               const float* __restrict__ av,
                               float* __restrict__ score, unsigned* __restrict__ mkey)
{
    int e = blockIdx.x * blockDim.x + threadIdx.x;
    if (e >= E + N) return;
    int s, d;
    if (e < E) { s = src[e]; d = dst[e]; } else { s = d = e - E; }  // self loop
    const float4* pl = (const float4*)(xl + (long)s * F);
    const float4* pr = (const float4*)(xr + (long)d * F);
    const float4* pa = (const float4*)av;
    float acc = 0.f;
#pragma unroll 4
    for (int i = 0; i < F / 4; ++i) {
        float4 l = pl[i], r = pr[i], a = pa[i];
        float z;
        z = l.x + r.x; acc += (z > 0.f ? z : NEG_SLOPE * z) * a.x;
        z = l.y + r.y; acc += (z > 0.f ? z : NEG_SLOPE * z) * a.y;
        z = l.z + r.z; acc += (z > 0.f ? z : NEG_SLOPE * z) * a.z;
        z = l.w + r.w; acc += (z > 0.f ? z : NEG_SLOPE * z) * a.w;
    }
    score[e] = acc;
    atomicMax(mkey + d, f2key(acc));
}

// pass 2: denom[dst] += exp(score - m[dst])
__global__ void edge_denom(const int* __restrict__ dst, int E, int N,
                           const float* __restrict__ score,
                           const unsigned* __restrict__ mkey,
                           float* __restrict__ denom)
{
    int e = blockIdx.x * blockDim.x + threadIdx.x;
    if (e >= E + N) return;
    int d = (e < E) ? dst[e] : e - E;
    atomicAdd(denom + d, __expf(score[e] - key2f(mkey[d])));
}

// pass 3: out[dst] += alpha * xl[src]   (one thread per edge x 4-feature chunk)
template<int F>
__global__ void edge_aggregate(const int* __restrict__ src, const int* __restrict__ dst,
                               int E, int N,
                               const float* __restrict__ score,
                               const unsigned* __restrict__ mkey,
                               const float* __restrict__ denom,
                               const float* __restrict__ xl,
                               float* __restrict__ out)
{
    constexpr int CH = F / 4;
    long t = (long)blockIdx.x * blockDim.x + threadIdx.x;
    long total = (long)(E + N) * CH;
    if (t >= total) return;
    int e = (int)(t / CH);                 // CH is power of two -> shift
    int c = (int)(t & (CH - 1));
    int s, d;
    if (e < E) { s = src[e]; d = dst[e]; } else { s = d = e - E; }
    float alpha = __expf(score[e] - key2f(mkey[d])) / denom[d];
    float4 v = ((const float4*)(xl + (long)s * F))[c];
    float* o = out + (long)d * F + (long)c * 4;
    atomicAdd(o + 0, alpha * v.x);
    atomicAdd(o + 1, alpha * v.y);
    atomicAdd(o + 2, alpha * v.z);
    atomicAdd(o + 3, alpha * v.w);
}

// relu + eval-mode BatchNorm (running stats mean=0, var=1), in place
template<int F>
__global__ void relu_bn(float* __restrict__ h, const float* __restrict__ g,
                        const float* __restrict__ be, long total)
{
    long i = (long)blockIdx.x * blockDim.x + threadIdx.x;
    if (i >= total) return;
    int f = (int)(i & (F - 1));
    float v = h[i];
    v = v > 0.f ? v : 0.f;
    h[i] = v * (g[f] * rsqrtf(1.f + BN_EPS)) + be[f];
}

// pooled[batch[n], colOff + f] += h[n, f]  (DUP>=0: also duplicated at DUP)
template<int F, int COL, int DUP>
__global__ void pool_add(const float* __restrict__ h, const int* __restrict__ batch,
                         int N, float* __restrict__ pool)
{
    long i = (long)blockIdx.x * blockDim.x + threadIdx.x;
    if (i >= (long)N * F) return;
    int n = (int)(i / F);
    int f = (int)(i & (F - 1));
    float v = h[i];
    long row = (long)batch[n] * POOL_COLS;
    atomicAdd(pool + row + COL + f, v);
    if (DUP >= 0) atomicAdd(pool + row + DUP + f, v);
}

// lin1 epilogue: t = bn5(relu(t + b))
__global__ void lin1_post(float* __restrict__ t, const float* __restrict__ b,
                          const float* __restrict__ g, const float* __restrict__ be,
                          long total)
{
    long i = (long)blockIdx.x * blockDim.x + threadIdx.x;
    if (i >= total) return;
    int f = (int)(i & 127);
    float v = t[i] + b[f];
    v = v > 0.f ? v : 0.f;
    t[i] = v * (g[f] * rsqrtf(1.f + BN_EPS)) + be[f];
}

// final head: logits += b2, emit sigmoid [512,16] then log_softmax [512,16]
__global__ void head_out(const float* __restrict__ logits, const float* __restrict__ b2,
                         float* __restrict__ out)
{
    int g = blockIdx.x * blockDim.x + threadIdx.x;
    if (g >= NGRAPH) return;
    float l[16];
    float mx = -3.0e38f;
#pragma unroll
    for (int j = 0; j < 16; ++j) {
        l[j] = logits[g * 16 + j] + b2[j];
        mx = fmaxf(mx, l[j]);
    }
    float se = 0.f;
#pragma unroll
    for (int j = 0; j < 16; ++j) se += __expf(l[j] - mx);
    float lse = mx + __logf(se);
#pragma unroll
    for (int j = 0; j < 16; ++j) {
        out[g * 16 + j]               = 1.f / (1.f + __expf(-l[j]));
        out[NGRAPH * 16 + g * 16 + j] = l[j] - lse;
    }
}

// ------------------------- host orchestration -------------------------------
static inline int cdiv(long a, long b) { return (int)((a + b - 1) / b); }

template<int FIN, int FOUT>
static void run_gat_layer(const float* feat,
                          const float* Wl, const float* Wr, const float* av,
                          const float* b, const float* g, const float* be,
                          const int* src, const int* dst, int E, int N,
                          float* xl, float* xr, float* score,
                          unsigned* mkey, float* denom, float* hout,
                          hipStream_t stream)
{
    const int ET = E + N;
    // xl = feat@Wl, xr = feat@Wr : one wave per 16x32 strip (NT=2, dual-B)
    long waves = (long)(N / 16) * (FOUT / 32);
    wmma_gemm<2, true><<<cdiv(waves * 32, 256), 256, 0, stream>>>(
        feat, Wl, Wr, xl, xr, N, FIN, FOUT);
    fill_u32<<<cdiv(N, 256), 256, 0, stream>>>(mkey, 0u, N);
    fill_f32<<<cdiv(N, 256), 256, 0, stream>>>(denom, 0.f, N);
    edge_score_max<FOUT><<<cdiv(ET, 256), 256, 0, stream>>>(
        src, dst, E, N, xl, xr, av, score, mkey);
    edge_denom<<<cdiv(ET, 256), 256, 0, stream>>>(dst, E, N, score, mkey, denom);
    long tot = (long)N * FOUT;
    init_bias<FOUT><<<cdiv(tot, 256), 256, 0, stream>>>(hout, b, tot);
    long agg = (long)ET * (FOUT / 4);
    edge_aggregate<FOUT><<<cdiv(agg, 256), 256, 0, stream>>>(
        src, dst, E, N, score, mkey, denom, xl, hout);
    relu_bn<FOUT><<<cdiv(tot, 256), 256, 0, stream>>>(hout, g, be, tot);
}

extern "C" void kernel_launch(void* const* d_in, const int* in_sizes, int n_in,
                              void* d_out, int out_size, void* d_ws, size_t ws_size,
                              hipStream_t stream)
{
    (void)n_in; (void)out_size; (void)ws_size;
    const float* x     = (const float*)d_in[0];
    const int*   ei    = (const int*)d_in[1];
    const int*   batch = (const int*)d_in[2];
    const int E = in_sizes[1] / 2;
    const int N = in_sizes[2];
    const int* src = ei;        // edge_index[0]
    const int* dst = ei + E;    // edge_index[1]

    // per-layer params: base = 4 + 6*li: Wl, Wr, a, b, g, be
    const float* P[4][6];
    for (int li = 0; li < 4; ++li)
        for (int j = 0; j < 6; ++j)
            P[li][j] = (const float*)d_in[4 + 6 * li + j];
    const float* lin1_W = (const float*)d_in[28];
    const float* lin1_b = (const float*)d_in[29];
    const float* g5     = (const float*)d_in[30];
    const float* be5    = (const float*)d_in[31];
    const float* lin2_W = (const float*)d_in[32];
    const float* lin2_b = (const float*)d_in[33];

    // workspace carve (floats)
    float* ws    = (float*)d_ws;
    float* xl    = ws;                          // N*128
    float* xr    = xl + (size_t)N * 128;        // N*128
    float* h1    = xr + (size_t)N * 128;        // N*128
    float* h2    = h1 + (size_t)N * 128;        // N*64
    float* h3    = h2 + (size_t)N * 64;         // N*32
    float* score = h3 + (size_t)N * 32;         // E+N
    unsigned* mkey = (unsigned*)(score + (size_t)(E + N));   // N
    float* denom = (float*)(mkey + N);          // N
    float* pool  = denom + N;                   // 512*256
    float* t1    = pool + (size_t)NGRAPH * POOL_COLS;        // 512*128
    float* logit = t1 + (size_t)NGRAPH * 128;                // 512*16

    // 3 live GATv2 layers (layer 4 is dead: h4 = h3 in the reference)
    run_gat_layer<128, 128>(x,  P[0][0], P[0][1], P[0][2], P[0][3], P[0][4], P[0][5],
                            src, dst, E, N, xl, xr, score, mkey, denom, h1, stream);
    run_gat_layer<128,  64>(h1, P[1][0], P[1][1], P[1][2], P[1][3], P[1][4], P[1][5],
                            src, dst, E, N, xl, xr, score, mkey, denom, h2, stream);
    run_gat_layer< 64,  32>(h2, P[2][0], P[2][1], P[2][2], P[2][3], P[2][4], P[2][5],
                            src, dst, E, N, xl, xr, score, mkey, denom, h3, stream);

    // pooling: concat [p1 | p2 | p3 | p4==p3] -> [512, 256]
    long pn = (long)NGRAPH * POOL_COLS;
    fill_f32<<<cdiv(pn, 256), 256, 0, stream>>>(pool, 0.f, pn);
    pool_add<128,   0,  -1><<<cdiv((long)N * 128, 256), 256, 0, stream>>>(h1, batch, N, pool);
    pool_add< 64, 128,  -1><<<cdiv((long)N *  64, 256), 256, 0, stream>>>(h2, batch, N, pool);
    pool_add< 32, 192, 224><<<cdiv((long)N *  32, 256), 256, 0, stream>>>(h3, batch, N, pool);

    // head: lin1 -> relu -> bn5 -> lin2 -> sigmoid / log_softmax
    long w1 = (long)(NGRAPH / 16) * (128 / 64);      // NT=4 -> 64-col strips
    wmma_gemm<4, false><<<cdiv(w1 * 32, 256), 256, 0, stream>>>(
        pool, lin1_W, nullptr, t1, nullptr, NGRAPH, 256, 128);
    lin1_post<<<cdiv((long)NGRAPH * 128, 256), 256, 0, stream>>>(
        t1, lin1_b, g5, be5, (long)NGRAPH * 128);
    long w2 = (long)(NGRAPH / 16);                   // NT=1, N=16
    wmma_gemm<1, false><<<cdiv(w2 * 32, 256), 256, 0, stream>>>(
        t1, lin2_W, nullptr, logit, nullptr, NGRAPH, 128, 16);
    head_out<<<cdiv(NGRAPH, 256), 256, 0, stream>>>(logit, lin2_b, (float*)d_out);
}